// OrientedRCNN_944892805765
// MI455X (gfx1250) — compile-verified
//
#include <hip/hip_runtime.h>

typedef float v2f __attribute__((ext_vector_type(2)));
typedef float v8f __attribute__((ext_vector_type(8)));

#define RS 7      // output grid size S
#define RC 256    // channels

// One workgroup per ROI. Waves 0..3 compute the 49 rotated grid coordinates
// via V_WMMA_F32_16X16X4_F32 (the genuine matmul in this op: [gx gy 1] x 3x2
// rotation+translation), results broadcast through LDS. Then each of the 256
// threads owns one channel and performs the memory-bound bilinear gather.
__global__ __launch_bounds__(256)
void rroi_align_kernel(const float* __restrict__ f0, const float* __restrict__ f1,
                       const float* __restrict__ f2, const float* __restrict__ f3,
                       const float* __restrict__ rois, const int* __restrict__ levels,
                       float* __restrict__ out, int N) {
  const int n   = blockIdx.x;
  const int b   = blockIdx.y;
  const int tid = threadIdx.x;
  const int roi = b * N + n;

  // ---- ROI parameters -------------------------------------------------
  const float* r = rois + (size_t)roi * 6;
  const float x = r[0], y = r[1], w = r[2], h = r[3], a = r[4], bb = r[5];

  // ---- parallelogram -> rectangle (per-ROI scalar math) ---------------
  const float v1x = x + a,        v1y = y + 0.5f * h;
  const float v2x = x + 0.5f * w, v2y = y + bb;
  const float v3x = x - a,        v3y = y - 0.5f * h;
  const float d1 = sqrtf(4.f * a * a + h * h);      // |v3 - v1|
  const float d2 = sqrtf(w * w + 4.f * bb * bb);    // |v4 - v2|
  const bool  cond = d1 > d2;
  const float sA = d1 / (d2 + 1e-6f);
  const float sB = d2 / (d1 + 1e-6f);
  const float d24x = 0.5f * w, d24y = bb;   // dir24
  const float d13x = a,        d13y = 0.5f * h;   // dir13
  const float V0x = cond ? v1x : v1x + (sB - 1.f) * d13x;
  const float V0y = cond ? v1y : v1y + (sB - 1.f) * d13y;
  const float V1x = cond ? v2x + (sA - 1.f) * d24x : v2x;
  const float V1y = cond ? v2y + (sA - 1.f) * d24y : v2y;
  const float V2x = cond ? v3x : v3x - (sB - 1.f) * d13x;
  const float V2y = cond ? v3y : v3y - (sB - 1.f) * d13y;
  const float ex = V1x - V0x, ey = V1y - V0y;
  const float gxv = V2x - V1x, gyv = V2y - V1y;
  const float h_rect = sqrtf(ex * ex + ey * ey);
  const float w_rect = sqrtf(gxv * gxv + gyv * gyv);
  float theta = atan2f(V1y - V2y, V1x - V2x);
  const float HPI = 1.5707963267948966f;
  theta = fminf(fmaxf(theta, -HPI), HPI);
  const float ct = cosf(theta), st = sinf(theta);

  // ---- level selection -------------------------------------------------
  const int   lvl   = levels[roi];
  const int   W     = 256 >> lvl;
  const int   H     = 256 >> lvl;
  const float scale = 0.25f / (float)(1 << lvl);
  const float* feat = (lvl == 0) ? f0 : (lvl == 1) ? f1 : (lvl == 2) ? f2 : f3;

  const float x_r = x * scale,      y_r = y * scale;
  const float w_r = w_rect * scale, h_r = h_rect * scale;
  // px = rx*scale*W/(W-1) - 0.5 (faithful to reference's double-scale unmap)
  const float fxs = scale * (float)W / (float)(W - 1);
  const float fys = scale * (float)H / (float)(H - 1);

  __shared__ float s_px[64];
  __shared__ float s_py[64];

  const int wave = tid >> 5;
  const int lane = tid & 31;
  if (wave < 4) {                    // uniform per wave -> EXEC all-1s inside
    const int m  = lane & 15;        // row within 16x16 tile
    const int p  = wave * 16 + m;    // grid point id (0..63, 49..63 padding)
    const int pc = p < 49 ? p : 48;
    const float tc = (float)(pc % 7) * (1.f / 6.f) - 0.5f;  // x-varying
    const float tr = (float)(pc / 7) * (1.f / 6.f) - 0.5f;  // y-varying
    const bool hi = lane >= 16;

    // A (16x4, MxK): row p = [gx, gy, 1, 0]
    // layout: VGPR0 = K0 (lanes 0-15) / K2 (lanes 16-31); VGPR1 = K1 / K3
    v2f A;
    A.x = hi ? 1.f : w_r * tc;
    A.y = hi ? 0.f : h_r * tr;

    // B (4x16, KxN): K0=[ct,-st,0..], K1=[st,ct,0..], K2=[x_r,y_r,0..], K3=0
    // layout: VGPR0 = K0 (lanes 0-15) / K2 (lanes 16-31); VGPR1 = K1 / K3
    v2f Bv;
    Bv.x = hi ? (m == 0 ? x_r : (m == 1 ? y_r : 0.f))
              : (m == 0 ? ct  : (m == 1 ? -st : 0.f));
    Bv.y = hi ? 0.f
              : (m == 0 ? st  : (m == 1 ? ct  : 0.f));

    v8f Dv = {0.f, 0.f, 0.f, 0.f, 0.f, 0.f, 0.f, 0.f};
    Dv = __builtin_amdgcn_wmma_f32_16x16x4_f32(
        /*neg_a=*/false, A, /*neg_b=*/false, Bv,
        /*c_mod=*/(short)0, Dv, /*reuse_a=*/false, /*reuse_b=*/false);

    // D layout: VGPR i -> (M=i, N=lane) for lanes 0-15; (M=i+8, N=lane-16) hi
    // column N=0 holds rx, N=1 holds ry  -> lanes 0,1,16,17 write to LDS
    if (m < 2) {
      const int mbase = wave * 16 + (hi ? 8 : 0);
#pragma unroll
      for (int i = 0; i < 8; ++i) {
        const float v = Dv[i];
        if (m == 0) s_px[mbase + i] = v * fxs - 0.5f;
        else        s_py[mbase + i] = v * fys - 0.5f;
      }
    }
  }
  __syncthreads();

  // ---- bilinear gather: one thread per channel ------------------------
  const int c = tid;
  const float* __restrict__ fb = feat + ((size_t)b * RC + c) * (size_t)(H * W);
  float* __restrict__ ob = out + ((size_t)roi * RC + c) * (RS * RS);

#pragma unroll 7
  for (int p = 0; p < RS * RS; ++p) {
    const float px = s_px[p], py = s_py[p];
    const float x0f = floorf(px), y0f = floorf(py);
    const float wx = px - x0f,  wy = py - y0f;   // weights use unclamped floor
    int x0 = (int)x0f; x0 = x0 < 0 ? 0 : (x0 > W - 1 ? W - 1 : x0);
    int x1 = x0 + 1;   x1 = x1 > W - 1 ? W - 1 : x1;
    int y0 = (int)y0f; y0 = y0 < 0 ? 0 : (y0 > H - 1 ? H - 1 : y0);
    int y1 = y0 + 1;   y1 = y1 > H - 1 ? H - 1 : y1;
    const float* r0 = fb + (size_t)y0 * W;
    const float* r1 = fb + (size_t)y1 * W;
    const float v00 = r0[x0], v01 = r0[x1];
    const float v10 = r1[x0], v11 = r1[x1];
    const float top = v00 + wx * (v01 - v00);
    const float bot = v10 + wx * (v11 - v10);
    ob[p] = top + wy * (bot - top);
  }
}

extern "C" void kernel_launch(void* const* d_in, const int* in_sizes, int n_in,
                              void* d_out, int out_size, void* d_ws, size_t ws_size,
                              hipStream_t stream) {
  (void)n_in; (void)out_size; (void)d_ws; (void)ws_size;
  const float* f0     = (const float*)d_in[0];
  const float* f1     = (const float*)d_in[1];
  const float* f2     = (const float*)d_in[2];
  const float* f3     = (const float*)d_in[3];
  const float* rois   = (const float*)d_in[4];
  const int*   levels = (const int*)d_in[5];
  float*       out    = (float*)d_out;

  // f0 = B*C*256*256 with C=256 -> B; levels = B*N -> N
  const int B = in_sizes[0] / (256 * 256 * 256);
  const int N = in_sizes[5] / (B > 0 ? B : 1);

  dim3 grid((unsigned)N, (unsigned)B);
  dim3 block(256);
  hipLaunchKernelGGL(rroi_align_kernel, grid, block, 0, stream,
                     f0, f1, f2, f3, rois, levels, out, N);
}